// SGSC_13245679140980
// MI455X (gfx1250) — compile-verified
//
#include <hip/hip_runtime.h>
#include <hip/hip_bf16.h>

typedef __attribute__((ext_vector_type(2))) float v2f;
typedef __attribute__((ext_vector_type(8))) float v8f;

#define B_    64
#define NROW  560
#define EMB   128
#define DM_   256
#define PN    14
#define NSWEEPS 8
#define LDAE  257
#define LDT   18      // LDS tile stride (even -> 8B-aligned v2f loads)
// dynamic LDS for jacobi: A(256*257) + c/s(256) + p/q(256 ints) + sort(512)
#define SMEM_J ((256*257 + 128*2 + 128*2 + 256*2) * 4)

// ---------------------------------------------------------------------------
// Generic batched fp32 GEMM via V_WMMA_F32_16X16X4_F32.
// C[b][m][n] = (accum? C : 0) + sum_k opA(A)[m,k]*B[k,n] + bias[n] + resid
// A element (m,k) at A + b*sAb + m*sAm + k*sAk   (strides in elements)
// B element (k,n) at B + b*sBb + k*sBk + n*sBn
// flags: bit0 = relu(A), bit1 = accumulate into C
// LDS: A tile row-major [m][k] stride 18; B tile stored transposed [n][k]
// stride 18 so both WMMA fragments are single aligned ds_load_b64's.
// ---------------------------------------------------------------------------
__global__ __launch_bounds__(256) void gemm_f32_wmma(
    const float* __restrict__ A, const float* __restrict__ B,
    const float* __restrict__ bias, const float* __restrict__ resid,
    float* __restrict__ C, int M, int N, int K, int batch,
    long long sAm, long long sAk, long long sAb,
    long long sBk, long long sBn, long long sBb,
    long long sCb, long long sRb, int flags)
{
    __shared__ __align__(16) float As[64 * LDT];   // [m][k]
    __shared__ __align__(16) float Bs[64 * LDT];   // [n][k] (transposed)
    const int b  = blockIdx.z;
    const int m0 = blockIdx.y * 64;
    const int n0 = blockIdx.x * 64;
    const float* Ab = A + (long long)b * sAb;
    const float* Bb = B + (long long)b * sBb;
    const int tid   = threadIdx.x;
    const int lane  = tid & 31;
    const int wid   = tid >> 5;     // 8 waves
    const int waveM = wid & 3;      // 4 m-tiles of 16
    const int waveN = wid >> 2;     // 2 n-groups of 32
    const bool reluA = (flags & 1) != 0;

    v8f acc0 = {};
    v8f acc1 = {};
    const int ktiles = (K + 15) >> 4;
    for (int kt = 0; kt < ktiles; ++kt) {
        const int k0 = kt << 4;
        // stage A 64x16
        #pragma unroll
        for (int r = 0; r < 4; ++r) {
            int idx = (r << 8) + tid;
            int mm = idx >> 4, kk = idx & 15;
            int gm = m0 + mm, gk = k0 + kk;
            float v = 0.0f;
            if (gm < M && gk < K) v = Ab[(long long)gm * sAm + (long long)gk * sAk];
            if (reluA) v = v > 0.0f ? v : 0.0f;
            As[mm * LDT + kk] = v;
        }
        // stage B 16x64, stored transposed [n][k]
        #pragma unroll
        for (int r = 0; r < 4; ++r) {
            int idx = (r << 8) + tid;
            int kk = idx >> 6, nn = idx & 63;
            int gk = k0 + kk, gn = n0 + nn;
            float v = 0.0f;
            if (gn < N && gk < K) v = Bb[(long long)gk * sBk + (long long)gn * sBn];
            Bs[nn * LDT + kk] = v;
        }
        __syncthreads();
        // ISA layouts: A 16x4 f32: lane L holds (M=L%16, K=2*(L/16)+i) in vgpr i
        //              B 4x16 f32: lane L holds (K=2*(L/16)+i, N=L%16) in vgpr i
        const int arow  = (waveM << 4) + (lane & 15);
        const int khalf = (lane >> 4) << 1;
        const int bc0   = (waveN << 5) + (lane & 15);
        #pragma unroll
        for (int k4 = 0; k4 < 4; ++k4) {
            const int kb = (k4 << 2) + khalf;           // even -> 8B aligned
            v2f a   = *(const v2f*)&As[arow * LDT + kb];
            v2f b0v = *(const v2f*)&Bs[bc0 * LDT + kb];
            v2f b1v = *(const v2f*)&Bs[(bc0 + 16) * LDT + kb];
            acc0 = __builtin_amdgcn_wmma_f32_16x16x4_f32(false, a, false, b0v,
                                                         (short)0, acc0, false, false);
            acc1 = __builtin_amdgcn_wmma_f32_16x16x4_f32(false, a, false, b1v,
                                                         (short)0, acc1, false, false);
        }
        __syncthreads();
    }
    // C layout: lane L holds (M = 8*(L/16)+i, N = L%16) in vgpr i
    float* Cb = C + (long long)b * sCb;
    const float* Rb = resid ? (resid + (long long)b * sRb) : nullptr;
    const bool accum = (flags & 2) != 0;
    const int rowhalf = (lane >> 4) << 3;
    #pragma unroll
    for (int j = 0; j < 2; ++j) {
        const int col = n0 + (waveN << 5) + (j << 4) + (lane & 15);
        v8f acc = j ? acc1 : acc0;
        if (col < N) {
            const float bv = bias ? bias[col] : 0.0f;
            #pragma unroll
            for (int i = 0; i < 8; ++i) {
                const int row = m0 + (waveM << 4) + rowhalf + i;
                if (row < M) {
                    long long offc = (long long)row * N + col;
                    float v = acc[i] + bv;
                    if (Rb) v += Rb[offc];
                    if (accum) v += Cb[offc];
                    Cb[offc] = v;
                }
            }
        }
    }
}

static void launch_gemm(hipStream_t s, const float* A, const float* B,
                        const float* bias, const float* resid, float* C,
                        int M, int N, int K, int batch,
                        long long sAm, long long sAk, long long sAb,
                        long long sBk, long long sBn, long long sBb,
                        long long sCb, long long sRb, int flags)
{
    dim3 grid((N + 63) / 64, (M + 63) / 64, batch);
    gemm_f32_wmma<<<grid, 256, 0, s>>>(A, B, bias, resid, C, M, N, K, batch,
                                       sAm, sAk, sAb, sBk, sBn, sBb, sCb, sRb, flags);
}

// ---------------------------------------------------------------------------
// Embed (x[b,:,n] @ tfw^T + tfb) fused with per-row RevIN norm over 128 embeds
// one WG (128 threads) per row (b,n). Stores xn, mean, stdev.
// ---------------------------------------------------------------------------
__global__ __launch_bounds__(128) void embed_ln_kernel(
    const float* __restrict__ x, const float* __restrict__ tfw,
    const float* __restrict__ tfb, const float* __restrict__ rw,
    const float* __restrict__ rb, float* __restrict__ xn,
    float* __restrict__ meanO, float* __restrict__ stdO)
{
    __shared__ float xcol[24];
    __shared__ float red[128];
    __shared__ float red2[128];
    const int row = blockIdx.x;          // b*560 + n
    const int b = row / NROW, n = row % NROW;
    const int e = threadIdx.x;
    if (e < 24) xcol[e] = x[((long long)b * 24 + e) * NROW + n];
    __syncthreads();
    float v = tfb[e];
    #pragma unroll
    for (int l = 0; l < 24; ++l) v += xcol[l] * tfw[e * 24 + l];
    red[e] = v; red2[e] = v * v;
    __syncthreads();
    for (int s = 64; s > 0; s >>= 1) {
        if (e < s) { red[e] += red[e + s]; red2[e] += red2[e + s]; }
        __syncthreads();
    }
    const float mean = red[0] * (1.0f / 128.0f);
    const float var  = red2[0] * (1.0f / 128.0f) - mean * mean;
    const float sd   = sqrtf(var + 1e-5f);
    const int p = n % PN;
    xn[(long long)row * EMB + e] = (v - mean) / sd * rw[p] + rb[p];
    if (e == 0) { meanO[row] = mean; stdO[row] = sd; }
}

// ---------------------------------------------------------------------------
// Batched 256x256 symmetric eigensolver: parallel-order two-sided Jacobi.
// One WG (1024 threads, 32 wave32 waves) per matrix. A resides in 263KB LDS
// (CDNA5: 320KB/WGP). V accumulates in global (L2-resident). Eigenpairs are
// sorted ascending (bitonic) to match jnp.linalg.eigh ordering.
// ---------------------------------------------------------------------------
__global__ __launch_bounds__(1024) void jacobi_eigh(
    const float* __restrict__ Ain, float* __restrict__ Vtmp,
    float* __restrict__ Vout, float* __restrict__ evout)
{
    extern __shared__ float sm[];
    float* Ash = sm;                         // 256*257
    float* pc  = sm + 256 * LDAE;            // 128
    float* ps  = pc + 128;                   // 128
    int*   pp  = (int*)(ps + 128);           // 128
    int*   pq  = pp + 128;                   // 128
    float* sd  = (float*)(pq + 128);         // 256
    int*   si  = (int*)(sd + 256);           // 256

    const int b   = blockIdx.x;
    const int tid = threadIdx.x;
    const float* Ab = Ain + (long long)b * 65536;
    float* Vb = Vtmp + (long long)b * 65536;

    for (int idx = tid; idx < 65536; idx += 1024) {
        int r = idx >> 8, c = idx & 255;
        Ash[r * LDAE + c] = 0.5f * (Ab[idx] + Ab[c * 256 + r]);   // symmetrize
        Vb[idx] = (r == c) ? 1.0f : 0.0f;
    }
    __syncthreads();

    for (int sweep = 0; sweep < NSWEEPS; ++sweep) {
        for (int r = 0; r < 255; ++r) {
            if (tid < 128) {                 // round-robin pairing, all disjoint
                int i = tid, p, q;
                if (i == 0) { p = 255; q = r % 255; }
                else { p = (r + i) % 255; q = (r + 255 - i) % 255; }
                if (p > q) { int t = p; p = q; q = t; }
                const float app = Ash[p * LDAE + p];
                const float aqq = Ash[q * LDAE + q];
                const float apq = Ash[p * LDAE + q];
                float c, s;
                if (fabsf(apq) > 1e-30f) {
                    float theta = 0.5f * (aqq - app) / apq;
                    float t = copysignf(1.0f, theta) /
                              (fabsf(theta) + sqrtf(1.0f + theta * theta));
                    c = rsqrtf(1.0f + t * t);
                    s = t * c;
                } else { c = 1.0f; s = 0.0f; }
                pc[i] = c; ps[i] = s; pp[i] = p; pq[i] = q;
            }
            __syncthreads();
            // rows: A <- J^T A
            #pragma unroll 4
            for (int w = 0; w < 32; ++w) {
                int idx = tid + (w << 10);
                int i = idx >> 8, col = idx & 255;
                int p = pp[i], q = pq[i];
                float c = pc[i], s = ps[i];
                float t1 = Ash[p * LDAE + col], t2 = Ash[q * LDAE + col];
                Ash[p * LDAE + col] = c * t1 - s * t2;
                Ash[q * LDAE + col] = s * t1 + c * t2;
            }
            __syncthreads();
            // cols: A <- A J ;  V <- V J
            #pragma unroll 4
            for (int w = 0; w < 32; ++w) {
                int idx = tid + (w << 10);
                int i = idx >> 8, m = idx & 255;
                int p = pp[i], q = pq[i];
                float c = pc[i], s = ps[i];
                float t1 = Ash[m * LDAE + p], t2 = Ash[m * LDAE + q];
                Ash[m * LDAE + p] = c * t1 - s * t2;
                Ash[m * LDAE + q] = s * t1 + c * t2;
                float v1 = Vb[m * 256 + p], v2 = Vb[m * 256 + q];
                Vb[m * 256 + p] = c * v1 - s * v2;
                Vb[m * 256 + q] = s * v1 + c * v2;
            }
            __syncthreads();
        }
    }

    if (tid < 256) { sd[tid] = Ash[tid * LDAE + tid]; si[tid] = tid; }
    __syncthreads();
    for (int ksz = 2; ksz <= 256; ksz <<= 1) {         // bitonic ascending
        for (int jj = ksz >> 1; jj > 0; jj >>= 1) {
            if (tid < 256) {
                int ixj = tid ^ jj;
                if (ixj > tid) {
                    bool up = ((tid & ksz) == 0);
                    float dv = sd[tid], dw = sd[ixj];
                    if ((dv > dw) == up) {
                        sd[tid] = dw; sd[ixj] = dv;
                        int t = si[tid]; si[tid] = si[ixj]; si[ixj] = t;
                    }
                }
            }
            __syncthreads();
        }
    }
    if (tid < 256) evout[(long long)b * 256 + tid] = sd[tid];
    for (int idx = tid; idx < 65536; idx += 1024) {
        int m = idx >> 8, j = idx & 255;
        Vout[(long long)b * 65536 + m * 256 + j] = Vb[m * 256 + si[j]];
    }
}

// ---------- small fused elementwise / reduction kernels ----------
__global__ void mscale_kernel(const float* __restrict__ D, const float* __restrict__ ev,
                              float* __restrict__ T1, long long total)
{
    long long idx = (long long)blockIdx.x * blockDim.x + threadIdx.x;
    if (idx >= total) return;
    int b = (int)(idx >> 16);
    int j = (int)(idx & 255);
    float l = ev[b * 256 + j];
    l = l > 0.0f ? l : 0.01f * l;               // LeakyReLU(0.01)
    float linv = rsqrtf(l + 0.1f) * 0.02f;
    T1[idx] = D[idx] * linv;
}

__global__ void nanfix_kernel(float* __restrict__ A, long long total)
{
    long long idx = (long long)blockIdx.x * blockDim.x + threadIdx.x;
    if (idx >= total) return;
    float v = A[idx];
    if (v != v) v = 0.0f;
    else if (isinf(v)) v = v > 0.0f ? 1e10f : -1e10f;
    A[idx] = v;
}

__global__ __launch_bounds__(256) void wd_kernel(const float* __restrict__ M,
                                                 const float* __restrict__ V2,
                                                 float* __restrict__ Wd)
{
    int b = blockIdx.x, i = threadIdx.x;
    const float* Mr = M + (long long)b * 65536 + (long long)i * 256;
    const float* Vc = V2 + (long long)b * 65536 + i;
    float s = 0.0f;
    for (int j = 0; j < 256; ++j) s += Mr[j] * Vc[(long long)j * 256];
    Wd[b * 256 + i] = 0.02f * s;
}

__global__ void pmul_kernel(const float* __restrict__ X2, const float* __restrict__ Wd,
                            float* __restrict__ P, long long total)
{
    long long idx = (long long)blockIdx.x * blockDim.x + threadIdx.x;
    if (idx >= total) return;
    int j = (int)(idx & 255);
    int b = (int)(idx / ((long long)NROW * DM_));
    P[idx] = X2[idx] * Wd[b * 256 + j];
}

__global__ __launch_bounds__(256) void softmax_kernel(const float* __restrict__ P,
                                                      float* __restrict__ prr)
{
    __shared__ float red[256];
    const long long row = blockIdx.x;
    const int j = threadIdx.x;
    float v = P[row * 256 + j];
    red[j] = v; __syncthreads();
    for (int s = 128; s > 0; s >>= 1) {
        if (j < s) red[j] = fmaxf(red[j], red[j + s]);
        __syncthreads();
    }
    float mx = red[0]; __syncthreads();
    float ev = expf(v - mx);
    red[j] = ev; __syncthreads();
    for (int s = 128; s > 0; s >>= 1) {
        if (j < s) red[j] += red[j + s];
        __syncthreads();
    }
    prr[row * 256 + j] = ev / red[0];
}

__global__ __launch_bounds__(256) void wred_kernel(const float* __restrict__ prr,
                                                   const float* __restrict__ w1,
                                                   const float* __restrict__ w2,
                                                   float* __restrict__ w10,
                                                   float* __restrict__ w20)
{
    int b = blockIdx.x, p = threadIdx.x;
    float s1 = 0.0f, s2 = 0.0f;
    for (int n = 0; n < NROW; ++n) {
        float pr = prr[((long long)b * NROW + n) * 256 + p];
        s1 += pr * w1[n * 256 + p];
        s2 += pr * w2[n * 256 + p];
    }
    w10[b * 256 + p] = s1;
    w20[b * 256 + p] = s2;
}

__global__ void o2_kernel(const float* __restrict__ oa, const float* __restrict__ w10,
                          const float* __restrict__ w20, const float* __restrict__ b1,
                          const float* __restrict__ b2, float* __restrict__ o2,
                          long long total)
{
    long long idx = (long long)blockIdx.x * blockDim.x + threadIdx.x;
    if (idx >= total) return;
    int b = (int)(idx >> 16);
    int j = (int)(idx & 255);
    float o1 = oa[idx] * w10[b * 256 + j] - b1[j];
    o2[idx] = o1 * w20[b * 256 + j] - b2[j];
}

__global__ void denorm_kernel(float* __restrict__ x1, const float* __restrict__ rw,
                              const float* __restrict__ rb, const float* __restrict__ meanI,
                              const float* __restrict__ stdI, long long total)
{
    long long idx = (long long)blockIdx.x * blockDim.x + threadIdx.x;
    if (idx >= total) return;
    long long row = idx >> 7;               // /128
    int p = (int)(row % PN);                // 560 % 14 == 0
    float v = x1[idx];
    v = (v - rb[p]) / (rw[p] + 1e-10f);
    x1[idx] = v * stdI[row] + meanI[row];
}

// ---------------------------------------------------------------------------
extern "C" void kernel_launch(void* const* d_in, const int* in_sizes, int n_in,
                              void* d_out, int out_size, void* d_ws, size_t ws_size,
                              hipStream_t stream)
{
    (void)in_sizes; (void)n_in; (void)out_size; (void)ws_size;
    const float* x    = (const float*)d_in[0];
    const float* tfw  = (const float*)d_in[1];
    const float* tfb  = (const float*)d_in[2];
    const float* f21w = (const float*)d_in[3];
    const float* f21b = (const float*)d_in[4];
    const float* f22w = (const float*)d_in[5];
    const float* f22b = (const float*)d_in[6];
    const float* f23w = (const float*)d_in[7];
    const float* f23b = (const float*)d_in[8];
    const float* f24w = (const float*)d_in[9];
    const float* f24b = (const float*)d_in[10];
    const float* b1   = (const float*)d_in[11];   // use row 0
    const float* b2   = (const float*)d_in[12];   // use row 0
    const float* gcw  = (const float*)d_in[13];
    const float* gcb  = (const float*)d_in[14];
    const float* f12w = (const float*)d_in[15];
    const float* f12b = (const float*)d_in[16];
    const float* rw   = (const float*)d_in[17];
    const float* rb   = (const float*)d_in[18];
    float* out = (float*)d_out;

    float* ws = (float*)d_ws;
    size_t off = 0;
    auto alloc = [&](size_t n) { float* p = ws + off; off += (n + 63) & ~(size_t)63; return p; };

    const long long BN  = (long long)B_ * NROW;          // 35840 rows
    const long long BNE = BN * EMB;                      // [B,560,128]
    const long long BND = BN * DM_;                      // [B,560,256]
    const long long BDD = (long long)B_ * 65536;         // [B,256,256]

    float* XN   = alloc(BNE);
    float* MEAN = alloc(BN);
    float* STD  = alloc(BN);
    float* X1b  = alloc(BND);      // later reused as P
    float* X2b  = alloc(BND);      // later reused as ob
    float* XTX  = alloc(BDD);      // later XtEEtX, then oa
    float* EVEC = alloc(BDD);
    float* EV   = alloc((size_t)B_ * 256);
    float* MBUF = alloc(BDD);
    float* T1   = alloc(BDD);      // also Jacobi Vtmp, also T2
    float* RES  = alloc(BDD);
    float* V2   = alloc(BDD);
    float* ETX  = alloc(BDD);      // later o2
    float* EETX = alloc(BND);      // later prr, then x1
    float* W1   = alloc((size_t)NROW * DM_);
    float* W2   = alloc((size_t)NROW * DM_);
    float* WD   = alloc((size_t)B_ * 256);
    float* W10  = alloc((size_t)B_ * 256);
    float* W20  = alloc((size_t)B_ * 256);

    (void)hipFuncSetAttribute((const void*)jacobi_eigh,
                              hipFuncAttributeMaxDynamicSharedMemorySize, SMEM_J);

    // 1. embed + RevIN norm
    embed_ln_kernel<<<(int)BN, 128, 0, stream>>>(x, tfw, tfb, rw, rb, XN, MEAN, STD);

    // 2-3. X1 = xn @ f21^T + b ; X2 = xn @ f22^T + b
    launch_gemm(stream, XN, f21w, f21b, nullptr, X1b, NROW, DM_, EMB, B_,
                EMB, 1, (long long)NROW * EMB, 1, EMB, 0, (long long)NROW * DM_, 0, 0);
    launch_gemm(stream, XN, f22w, f22b, nullptr, X2b, NROW, DM_, EMB, B_,
                EMB, 1, (long long)NROW * EMB, 1, EMB, 0, (long long)NROW * DM_, 0, 0);

    // 4. XtX = X1^T X1
    launch_gemm(stream, X1b, X1b, nullptr, nullptr, XTX, DM_, DM_, NROW, B_,
                1, DM_, (long long)NROW * DM_, DM_, 1, (long long)NROW * DM_, 65536, 0, 0);

    // 5. eigh #1
    jacobi_eigh<<<B_, 1024, SMEM_J, stream>>>(XTX, T1, EVEC, EV);

    // 6. T1 = D * diag(linv) ; 7. M = T1 @ D^T  (symmetric)
    mscale_kernel<<<(int)((BDD + 255) / 256), 256, 0, stream>>>(EVEC, EV, T1, BDD);
    launch_gemm(stream, T1, EVEC, nullptr, nullptr, MBUF, DM_, DM_, DM_, B_,
                DM_, 1, 65536, 1, DM_, 65536, 65536, 0, 0);

    // 8-9. RES = M @ (XtX @ M)
    launch_gemm(stream, XTX, MBUF, nullptr, nullptr, T1, DM_, DM_, DM_, B_,
                DM_, 1, 65536, DM_, 1, 65536, 65536, 0, 0);
    launch_gemm(stream, MBUF, T1, nullptr, nullptr, RES, DM_, DM_, DM_, B_,
                DM_, 1, 65536, DM_, 1, 65536, 65536, 0, 0);

    // 10. EtX = relu(X1)^T X1 ; 11. EEtX = relu(X1) @ EtX ; 12. XtEEtX -> XTX
    launch_gemm(stream, X1b, X1b, nullptr, nullptr, ETX, DM_, DM_, NROW, B_,
                1, DM_, (long long)NROW * DM_, DM_, 1, (long long)NROW * DM_, 65536, 0, 1);
    launch_gemm(stream, X1b, ETX, nullptr, nullptr, EETX, NROW, DM_, DM_, B_,
                DM_, 1, (long long)NROW * DM_, DM_, 1, 65536, (long long)NROW * DM_, 0, 1);
    launch_gemm(stream, X1b, EETX, nullptr, nullptr, XTX, DM_, DM_, NROW, B_,
                1, DM_, (long long)NROW * DM_, DM_, 1, (long long)NROW * DM_, 65536, 0, 0);

    // 13-14. RES += M @ (XtEEtX @ M)
    launch_gemm(stream, XTX, MBUF, nullptr, nullptr, T1, DM_, DM_, DM_, B_,
                DM_, 1, 65536, DM_, 1, 65536, 65536, 0, 0);
    launch_gemm(stream, MBUF, T1, nullptr, nullptr, RES, DM_, DM_, DM_, B_,
                DM_, 1, 65536, DM_, 1, 65536, 65536, 0, 2);

    // 15. nan_to_num ; 16. eigh #2
    nanfix_kernel<<<(int)((BDD + 255) / 256), 256, 0, stream>>>(RES, BDD);
    jacobi_eigh<<<B_, 1024, SMEM_J, stream>>>(RES, T1, V2, EV);

    // 17. Wd = 0.02 * diag(M @ V2) ; 18. P = X2 * Wd (into X1 buffer)
    wd_kernel<<<B_, 256, 0, stream>>>(MBUF, V2, WD);
    pmul_kernel<<<(int)((BND + 255) / 256), 256, 0, stream>>>(X2b, WD, X1b, BND);

    // 19. w1/w2 from batch-0 rows
    launch_gemm(stream, XN, f23w, f23b, nullptr, W1, NROW, DM_, EMB, 1,
                EMB, 1, 0, 1, EMB, 0, 0, 0, 0);
    launch_gemm(stream, XN, f24w, f24b, nullptr, W2, NROW, DM_, EMB, 1,
                EMB, 1, 0, 1, EMB, 0, 0, 0, 0);

    // 20. oa = X2^T P (into XTX)
    launch_gemm(stream, X2b, X1b, nullptr, nullptr, XTX, DM_, DM_, NROW, B_,
                1, DM_, (long long)NROW * DM_, DM_, 1, (long long)NROW * DM_, 65536, 0, 0);

    // 21-22. prr = softmax(P) (into EETX) ; weighted diag reductions
    softmax_kernel<<<(int)BN, 256, 0, stream>>>(X1b, EETX);
    wred_kernel<<<B_, 256, 0, stream>>>(EETX, W1, W2, W10, W20);

    // 23. o2 = (oa*w10 - b1)*w20 - b2 (into ETX)
    o2_kernel<<<(int)((BDD + 255) / 256), 256, 0, stream>>>(XTX, W10, W20, b1, b2, ETX, BDD);

    // 24. ob = P @ o2 (into X2 buffer)
    launch_gemm(stream, X1b, ETX, nullptr, nullptr, X2b, NROW, DM_, DM_, B_,
                DM_, 1, (long long)NROW * DM_, DM_, 1, 65536, (long long)NROW * DM_, 0, 0);

    // 25. x1 = ob @ gcw^T + gcb + xn (into EETX)
    launch_gemm(stream, X2b, gcw, gcb, XN, EETX, NROW, EMB, DM_, B_,
                DM_, 1, (long long)NROW * DM_, 1, DM_, 0,
                (long long)NROW * EMB, (long long)NROW * EMB, 0);

    // 26. RevIN denorm in place
    denorm_kernel<<<(int)((BNE + 255) / 256), 256, 0, stream>>>(EETX, rw, rb, MEAN, STD, BNE);

    // 27. out = x1.reshape(2560,1792) @ f12w^T + f12b
    launch_gemm(stream, EETX, f12w, f12b, nullptr, out, B_ * 40, 768, PN * EMB, 1,
                (long long)PN * EMB, 1, 0, 1, (long long)PN * EMB, 0, 0, 0, 0);
}